// Query2Context_56727928046495
// MI455X (gfx1250) — compile-verified
//
#include <hip/hip_runtime.h>

typedef __attribute__((ext_vector_type(2))) float v2f;
typedef __attribute__((ext_vector_type(4))) float v4f;
typedef __attribute__((ext_vector_type(8))) float v8f;

constexpr int B = 32, T = 2048, J = 64, D = 256;
constexpr int TCHUNK = 128;
constexpr int NCHUNK = T / TCHUNK;   // 16

// ---------------- Kernel 1: z1[b,t] = max_j s[b,t,j] ----------------
__global__ __launch_bounds__(256) void k_z1(const float* __restrict__ s,
                                            float* __restrict__ z1) {
  int i = blockIdx.x * 256 + threadIdx.x;            // flat (b*T + t)
  const v4f* row = (const v4f*)(s + (size_t)i * J);  // 64 contiguous floats
  float m = -3.402823466e38f;
#pragma unroll
  for (int k = 0; k < J / 4; ++k) {
    v4f v = row[k];
    m = fmaxf(m, fmaxf(fmaxf(v.x, v.y), fmaxf(v.z, v.w)));
  }
  z1[i] = m;
}

// ---------------- Kernel 2: w[b,:] = softmax_T(z1[b,:]) ----------------
__global__ __launch_bounds__(256) void k_softmax(const float* __restrict__ z1,
                                                 float* __restrict__ w) {
  __shared__ float red[256];
  const int b = blockIdx.x, tid = threadIdx.x;
  const float* row = z1 + (size_t)b * T;
  float vals[T / 256];
  float m = -3.402823466e38f;
#pragma unroll
  for (int k = 0; k < T / 256; ++k) {
    vals[k] = row[tid + k * 256];
    m = fmaxf(m, vals[k]);
  }
  red[tid] = m;
  __syncthreads();
  for (int off = 128; off > 0; off >>= 1) {
    if (tid < off) red[tid] = fmaxf(red[tid], red[tid + off]);
    __syncthreads();
  }
  const float gmax = red[0];
  __syncthreads();
  float e[T / 256];
  float lsum = 0.f;
#pragma unroll
  for (int k = 0; k < T / 256; ++k) {
    e[k] = __expf(vals[k] - gmax);
    lsum += e[k];
  }
  red[tid] = lsum;
  __syncthreads();
  for (int off = 128; off > 0; off >>= 1) {
    if (tid < off) red[tid] += red[tid + off];
    __syncthreads();
  }
  const float inv = 1.0f / red[0];
#pragma unroll
  for (int k = 0; k < T / 256; ++k)
    w[(size_t)b * T + tid + k * 256] = e[k] * inv;
}

// -------- Kernel 3: chunk partials of pooled via V_WMMA_F32_16X16X4_F32 --------
// part[b, chunk, d] = sum_{t in chunk} w[b,t] * h[b,t,d]
__global__ __launch_bounds__(256) void k_pool(const float* __restrict__ h,
                                              const float* __restrict__ w,
                                              float* __restrict__ part) {
  const int chunk = blockIdx.x % NCHUNK;
  const int b     = blockIdx.x / NCHUNK;
  const int t0    = chunk * TCHUNK;
  const int wave  = threadIdx.x >> 5;
  const int lane  = threadIdx.x & 31;
  const int hi    = (lane >= 16) ? 1 : 0;  // half-wave selects K-pair {0,1} vs {2,3}
  const int n     = lane & 15;             // column within 16-wide tile
  const int d0    = wave * 32;             // each wave owns 32 d-columns (2 tiles)

  const float* hb  = h + ((size_t)b * T + t0) * D;
  const v4f*   wb4 = (const v4f*)(w + (size_t)b * T + t0);

  v8f c0 = {0.f, 0.f, 0.f, 0.f, 0.f, 0.f, 0.f, 0.f};
  v8f c1 = c0;

  for (int t = 0; t < TCHUNK; t += 4) {
    // A (16x4 f32): all 16 M-rows identical = w[t..t+3].
    // ISA layout: VGPR0 = K0 (lanes 0-15) | K2 (lanes 16-31); VGPR1 = K1 | K3.
    v4f ww = wb4[t >> 2];
    v2f a;
    a.x = hi ? ww.z : ww.x;
    a.y = hi ? ww.w : ww.y;
    // B (4x16 f32): VGPR0 = rows K0|K2, VGPR1 = rows K1|K3, N striped over lanes.
    const float* r0 = hb + (size_t)(t + 2 * hi) * D;  // row t+0 / t+2
    const float* r1 = r0 + D;                         // row t+1 / t+3
    v2f b0, b1;
    b0.x = r0[d0 + n];
    b0.y = r1[d0 + n];
    b1.x = r0[d0 + 16 + n];
    b1.y = r1[d0 + 16 + n];
    c0 = __builtin_amdgcn_wmma_f32_16x16x4_f32(false, a, false, b0, (short)0, c0,
                                               false, false);
    c1 = __builtin_amdgcn_wmma_f32_16x16x4_f32(false, a, false, b1, (short)0, c1,
                                               false, false);
  }
  // C layout: VGPR0, lanes 0-15 => (M=0, N=lane). All M-rows identical; take M=0.
  if (lane < 16) {
    float* dst = part + ((size_t)b * NCHUNK + chunk) * D + d0;
    dst[lane]      = c0[0];
    dst[16 + lane] = c1[0];
  }
}

// ---------------- Kernel 4: pooled[b,d] = sum_chunks part ----------------
__global__ __launch_bounds__(256) void k_reduce(const float* __restrict__ part,
                                                float* __restrict__ pooled) {
  const int b = blockIdx.x, d = threadIdx.x;
  float acc = 0.f;
#pragma unroll
  for (int c = 0; c < NCHUNK; ++c)
    acc += part[((size_t)b * NCHUNK + c) * D + d];
  pooled[(size_t)b * D + d] = acc;
}

// ---------------- Kernel 5: out[b,t,:] = pooled[b,:] ----------------
__global__ __launch_bounds__(256) void k_bcast(const float* __restrict__ pooled,
                                               float* __restrict__ out) {
  size_t i = (size_t)blockIdx.x * 256 + threadIdx.x;  // float4 slot in [B*T*D/4]
  int dv    = (int)(i & (D / 4 - 1));                 // which float4 within D
  size_t bt = i >> 6;                                 // (b*T + t)
  int b     = (int)(bt >> 11);                        // / T
  v4f p = ((const v4f*)(pooled + (size_t)b * D))[dv];
  ((v4f*)out)[i] = p;
}

extern "C" void kernel_launch(void* const* d_in, const int* in_sizes, int n_in,
                              void* d_out, int out_size, void* d_ws, size_t ws_size,
                              hipStream_t stream) {
  const float* h = (const float*)d_in[0];  // [B,T,D]
  const float* s = (const float*)d_in[1];  // [B,T,J]
  float* out = (float*)d_out;              // [B,T,D]

  float* ws     = (float*)d_ws;
  float* z1     = ws;                          // B*T
  float* w      = z1 + (size_t)B * T;          // B*T
  float* part   = w + (size_t)B * T;           // B*NCHUNK*D
  float* pooled = part + (size_t)B * NCHUNK * D;  // B*D

  k_z1<<<(B * T) / 256, 256, 0, stream>>>(s, z1);
  k_softmax<<<B, 256, 0, stream>>>(z1, w);
  k_pool<<<B * NCHUNK, 256, 0, stream>>>(h, w, part);
  k_reduce<<<B, 256, 0, stream>>>(part, pooled);
  k_bcast<<<(B * T * (D / 4)) / 256, 256, 0, stream>>>(pooled, out);
}